// linear_trunc_ind_3762391352094
// MI455X (gfx1250) — compile-verified
//
#include <hip/hip_runtime.h>
#include <stdint.h>

typedef __attribute__((ext_vector_type(2))) float v2f;
typedef __attribute__((ext_vector_type(8))) float v8f;

#define B_DIM 256
#define O_DIM 512
#define I_DIM 1024
#define K_TOP 10
#define PSTR  1028                       // padded LDS row stride (floats): conflict-free b64/b128 access
#define XS_OFF 0
#define WS_OFF (16 * PSTR)
#define CRED_OFF (32 * PSTR)
#define SMEM_FLOATS (32 * PSTR + 8 * 256)

// ---- gfx1250 async global->LDS staging (guarded; falls back to VGPR round-trip) ----
#if defined(__HIP_DEVICE_COMPILE__) && __has_builtin(__builtin_amdgcn_global_load_async_to_lds_b128)
#define HAS_ASYNC_LDS 1
typedef int v4i __attribute__((vector_size(16)));
typedef __attribute__((address_space(1))) v4i g1_v4i;   // global (AS1) 16B vector
typedef __attribute__((address_space(3))) v4i l3_v4i;   // LDS    (AS3) 16B vector

__device__ __forceinline__ void async_copy_b128(const void* gsrc, void* ldst) {
    // AS1 pointer: same numeric value as generic for global memory.
    // AS3 pointer: low 32 bits of a generic LDS pointer are the LDS byte offset.
    __builtin_amdgcn_global_load_async_to_lds_b128(
        (g1_v4i*)(uintptr_t)gsrc,
        (l3_v4i*)(uint32_t)(uintptr_t)ldst,
        0, 0);
}

__device__ __forceinline__ void async_wait_all() {
#if __has_builtin(__builtin_amdgcn_s_wait_asynccnt)
    __builtin_amdgcn_s_wait_asynccnt(0);
#else
    asm volatile("s_wait_asynccnt 0x0" ::: "memory");
#endif
}
#else
#define HAS_ASYNC_LDS 0
#endif

__device__ __forceinline__ void topk_update(float p, float (&t)[K_TOP], float (&s)[K_TOP]) {
    // t[0..9] descending (10 largest), s[0..9] ascending (10 smallest)
    if (p > t[K_TOP - 1]) {
        float v = p;
#pragma unroll
        for (int j = 0; j < K_TOP; ++j) {
            float mx = fmaxf(t[j], v);
            v = fminf(t[j], v);
            t[j] = mx;
        }
    }
    if (p < s[K_TOP - 1]) {
        float v = p;
#pragma unroll
        for (int j = 0; j < K_TOP; ++j) {
            float mn = fminf(s[j], v);
            v = fmaxf(s[j], v);
            s[j] = mn;
        }
    }
}

__global__ __launch_bounds__(256) void linear_trunc_fused_kernel(
    const float* __restrict__ x,       // (256, 1024)
    const float* __restrict__ w,       // (512, 1024)
    float* __restrict__ out)           // (256, 512)
{
    extern __shared__ float smem[];
    float* xs   = smem + XS_OFF;       // 16 rows of x-tile, stride PSTR
    float* ws   = smem + WS_OFF;       // 16 rows of w-tile, stride PSTR
    float* cred = smem + CRED_OFF;     // 8 waves x 16x16 partial GEMM tiles

    const int tid = threadIdx.x;
    const int b0  = blockIdx.y * 16;
    const int o0  = blockIdx.x * 16;

    // ---- Stage tiles into LDS: 16 rows x 256 float4 columns per tensor, coalesced ----
#pragma unroll
    for (int j = 0; j < 16; ++j) {
        int idx = tid + 256 * j;       // 0..4095
        int row = idx >> 8;
        int c4  = idx & 255;
        const float* gx = x + (size_t)(b0 + row) * I_DIM + c4 * 4;
        const float* gw = w + (size_t)(o0 + row) * I_DIM + c4 * 4;
        float* lx = xs + row * PSTR + c4 * 4;
        float* lw = ws + row * PSTR + c4 * 4;
#if HAS_ASYNC_LDS
        async_copy_b128(gx, lx);       // global_load_async_to_lds_b128 (ASYNCcnt)
        async_copy_b128(gw, lw);
#else
        *(float4*)lx = *(const float4*)gx;
        *(float4*)lw = *(const float4*)gw;
#endif
    }
#if HAS_ASYNC_LDS
    async_wait_all();                  // s_wait_asynccnt 0 before signaling the barrier
#endif
    __syncthreads();

    // ---- Partial GEMM on the matrix pipe: V_WMMA_F32_16X16X4_F32 ----
    // Wave `wave` covers K slice [wave*128, wave*128+128): 32 WMMA steps.
    const int wave = tid >> 5;
    const int lane = tid & 31;
    const int half = lane >> 4;        // 0: K pair (0,1); 1: K pair (2,3)
    const int r    = lane & 15;        // A: row M=r ; B: col N=r

    v8f c = {};
    {
        const float* arow = xs + r * PSTR;
        const float* brow = ws + r * PSTR;
        const int kbase = wave * 128 + 2 * half;
#pragma unroll 4
        for (int kk = 0; kk < 128; kk += 4) {
            v2f a = *(const v2f*)(arow + kbase + kk);
            v2f b = *(const v2f*)(brow + kbase + kk);
            c = __builtin_amdgcn_wmma_f32_16x16x4_f32(false, a, false, b,
                                                      (short)0, c, false, false);
        }
    }
    // C/D layout: element j of lane -> M = j + 8*half, N = r
#pragma unroll
    for (int j = 0; j < 8; ++j)
        cred[wave * 256 + (j + 8 * half) * 16 + r] = c[j];

    // ---- Per-pair top-10 / bottom-10 scan out of LDS ----
    const int bl = tid >> 4;
    const int ol = tid & 15;
    float t[K_TOP], s[K_TOP];
#pragma unroll
    for (int j = 0; j < K_TOP; ++j) { t[j] = -INFINITY; s[j] = INFINITY; }

    const float* xrow = xs + bl * PSTR;
    const float* wrow = ws + ol * PSTR;
    for (int i = 0; i < I_DIM; i += 4) {
        float4 xv = *(const float4*)(xrow + i);
        float4 wv = *(const float4*)(wrow + i);
        topk_update(xv.x * wv.x, t, s);
        topk_update(xv.y * wv.y, t, s);
        topk_update(xv.z * wv.z, t, s);
        topk_update(xv.w * wv.w, t, s);
    }
    __syncthreads();   // cred writes (all waves) complete before reduction reads

    // ---- Combine: out = gemm - sum(top10) + sum(bottom10) ----
    float gemm = 0.f;
#pragma unroll
    for (int w8 = 0; w8 < 8; ++w8)
        gemm += cred[w8 * 256 + bl * 16 + ol];

    float topSum = 0.f, botSum = 0.f;
#pragma unroll
    for (int j = 0; j < K_TOP; ++j) { topSum += t[j]; botSum += s[j]; }

    out[(size_t)(b0 + bl) * O_DIM + (o0 + ol)] = gemm - topSum + botSum;
}

extern "C" void kernel_launch(void* const* d_in, const int* in_sizes, int n_in,
                              void* d_out, int out_size, void* d_ws, size_t ws_size,
                              hipStream_t stream) {
    (void)in_sizes; (void)n_in; (void)d_ws; (void)ws_size; (void)out_size;
    const float* x = (const float*)d_in[0];
    const float* w = (const float*)d_in[1];
    // d_in[2] is k == 10, baked in as K_TOP.
    float* out = (float*)d_out;

    dim3 grid(O_DIM / 16, B_DIM / 16);   // 32 x 16 tiles
    dim3 block(256);
    size_t smem_bytes = (size_t)SMEM_FLOATS * sizeof(float);  // ~140 KB (CDNA5: 320 KB/WG)
    linear_trunc_fused_kernel<<<grid, block, smem_bytes, stream>>>(x, w, out);
}